// DisenConv_27693949125002
// MI455X (gfx1250) — compile-verified
//
#include <hip/hip_runtime.h>
#include <math.h>

// ---------------------------------------------------------------------------
// DisenConv (DisenGCN neighborhood routing) for MI455X / gfx1250.
//   z = l2norm(relu(X @ W + b))  -- f32 WMMA 16x16x4 GEMM (all 8 factors/wave)
//   6 routing iterations: edge softmax + fp32-atomic scatter-add + renormalize
// ---------------------------------------------------------------------------

#define INP 256
#define KF  8
#define DD  16
#define FDIM (KF * DD)   // 128
#define ROUTIT 6
#define EPS 1e-12f

typedef __attribute__((ext_vector_type(2))) float v2f;
typedef __attribute__((ext_vector_type(8))) float v8f;

// ---------------------------------------------------------------------------
// Kernel 1: projection GEMM + bias + relu + per-(n,k) L2 norm.
// One wave (32 threads) computes the full 16(row) x 128(K*D) output slab:
// the A fragment (rows of X) is loaded ONCE per K-step and fed to 8 WMMAs,
// one per factor. 8 x v8f accumulators = 64 VGPRs.
//   A tile (16x4 f32): lane lo = row, vgpr pair = K = 2*hi, 2*hi+1.
//   B tile (4x16 f32): lane lo = col, vgpr pair = K = 2*hi, 2*hi+1.
//   D tile: vgpr j <-> row j + 8*hi, lane lo <-> col.
// Writes z = c = agg = normalized result (agg seeds routing iteration 0).
// ---------------------------------------------------------------------------
__global__ __launch_bounds__(32)
void disen_init_gemm(const float* __restrict__ X,   // [N, 256]
                     const float* __restrict__ W,   // [8, 256, 16]
                     const float* __restrict__ B,   // [8, 16]
                     float* __restrict__ z,         // [N, 128]
                     float* __restrict__ c,         // [N, 128]
                     float* __restrict__ agg,       // [N, 128]
                     int N)
{
    const int lane = threadIdx.x;          // 0..31
    const int lo   = lane & 15;
    const int hi   = lane >> 4;            // 0 or 1
    const int rowbase = blockIdx.x * 16;

    // clamp load row so EXEC stays uniform through the WMMA loop
    int arow = rowbase + lo;
    if (arow >= N) arow = N - 1;

    const float* Xrow = X + (size_t)arow * INP;

    v8f acc[KF];
#pragma unroll
    for (int k = 0; k < KF; ++k) acc[k] = (v8f){};

    for (int kk = 0; kk < INP; kk += 4) {
        const int ka = kk + 2 * hi;
        // A: X[arow][ka..ka+1]  (contiguous, 8B aligned) -- shared by all k
        const float2 av = *(const float2*)(Xrow + ka);
        v2f A; A.x = av.x; A.y = av.y;
#pragma unroll
        for (int k = 0; k < KF; ++k) {
            const float* Wk = W + (size_t)k * INP * DD;
            v2f Bv;
            Bv.x = Wk[(size_t)ka * DD + lo];
            Bv.y = Wk[(size_t)(ka + 1) * DD + lo];
            acc[k] = __builtin_amdgcn_wmma_f32_16x16x4_f32(
                /*neg_a=*/false, A, /*neg_b=*/false, Bv,
                /*c_mod=*/(short)0, acc[k], /*reuse_a=*/false, /*reuse_b=*/false);
        }
    }

#pragma unroll
    for (int k = 0; k < KF; ++k) {
        const float bias = B[k * DD + lo];     // depends on column only
#pragma unroll
        for (int j = 0; j < 8; ++j) {
            float v = acc[k][j] + bias;
            v = v > 0.0f ? v : 0.0f;           // relu
            // L2 norm over the 16 columns of this row (stays inside half-wave)
            float ss = v * v;
            ss += __shfl_xor(ss, 1, 32);
            ss += __shfl_xor(ss, 2, 32);
            ss += __shfl_xor(ss, 4, 32);
            ss += __shfl_xor(ss, 8, 32);
            const float inv = 1.0f / fmaxf(sqrtf(ss), EPS);
            const float out = v * inv;

            const int row = rowbase + j + 8 * hi;
            if (row < N) {
                const size_t idx = (size_t)row * FDIM + k * DD + lo;
                z[idx]   = out;
                c[idx]   = out;
                agg[idx] = out;                // seed for routing iteration 0
            }
        }
    }
}

// ---------------------------------------------------------------------------
// Kernel 2: one wave32 per edge. Each lane owns a float4 (4 of 128 feats),
// 4 lanes per factor k (k = lane>>2). logits[k] = <z[src,k,:], c[trg,k,:]>,
// softmax over k, then agg[src] += p[k] * z[trg] via native fp32 atomics.
// ---------------------------------------------------------------------------
__global__ __launch_bounds__(256)
void disen_edge_route(const float* __restrict__ z,    // [N,128]
                      const float* __restrict__ c,    // [N,128]
                      float* __restrict__ agg,        // [N,128]
                      const int* __restrict__ src,
                      const int* __restrict__ trg,
                      int M)
{
    const int wave = (blockIdx.x * (blockDim.x >> 5)) + (threadIdx.x >> 5);
    if (wave >= M) return;                 // uniform within a wave
    const int lane = threadIdx.x & 31;

    const int s = src[wave];
    const int t = trg[wave];

    const float4* zs4 = (const float4*)(z + (size_t)s * FDIM);
    const float4* ct4 = (const float4*)(c + (size_t)t * FDIM);
    const float4* zt4 = (const float4*)(z + (size_t)t * FDIM);

    const float4 a  = zs4[lane];
    const float4 ct = ct4[lane];
    const float4 zt = zt4[lane];

    // partial dot for this lane's 4 features of factor k = lane>>2
    float dp = a.x * ct.x + a.y * ct.y + a.z * ct.z + a.w * ct.w;
    dp += __shfl_xor(dp, 1, 32);
    dp += __shfl_xor(dp, 2, 32);           // all 4 lanes of a k-group: logit_k (TAU=1)

    // softmax over k=8 (values replicated 4x per group; sum counts 4x)
    float mx = dp;
    mx = fmaxf(mx, __shfl_xor(mx, 4, 32));
    mx = fmaxf(mx, __shfl_xor(mx, 8, 32));
    mx = fmaxf(mx, __shfl_xor(mx, 16, 32));
    const float e = expf(dp - mx);
    float sm = e;
    sm += __shfl_xor(sm, 1, 32);
    sm += __shfl_xor(sm, 2, 32);
    sm += __shfl_xor(sm, 4, 32);
    sm += __shfl_xor(sm, 8, 32);
    sm += __shfl_xor(sm, 16, 32);          // = 4 * sum_k exp
    const float p = e * 4.0f / sm;

    float* dst = agg + (size_t)s * FDIM + lane * 4;
    unsafeAtomicAdd(dst + 0, p * zt.x);
    unsafeAtomicAdd(dst + 1, p * zt.y);
    unsafeAtomicAdd(dst + 2, p * zt.z);
    unsafeAtomicAdd(dst + 3, p * zt.w);
}

// ---------------------------------------------------------------------------
// Kernel 3: per (n,k) renormalize. Reads agg, writes:
//   c   = l2norm(agg)          (the routing state / final output)
//   agg = l2norm(agg)          (in-place; seeds the NEXT iteration's scatter)
//   z   = agg (raw)  if copy_z (torch in-place aliasing at t==0)
// Per-thread elements are disjoint, so the in-place agg update is safe.
// ---------------------------------------------------------------------------
__global__ __launch_bounds__(256)
void disen_renorm(float* __restrict__ agg,
                  float* __restrict__ c,
                  float* __restrict__ z,
                  int copy_z,
                  int NK)                  // N * 8
{
    const int tid = blockIdx.x * blockDim.x + threadIdx.x;
    if (tid >= NK) return;
    const size_t base = (size_t)tid * DD;

    float4 v0 = *(const float4*)(agg + base + 0);
    float4 v1 = *(const float4*)(agg + base + 4);
    float4 v2 = *(const float4*)(agg + base + 8);
    float4 v3 = *(const float4*)(agg + base + 12);

    float ss = v0.x*v0.x + v0.y*v0.y + v0.z*v0.z + v0.w*v0.w
             + v1.x*v1.x + v1.y*v1.y + v1.z*v1.z + v1.w*v1.w
             + v2.x*v2.x + v2.y*v2.y + v2.z*v2.z + v2.w*v2.w
             + v3.x*v3.x + v3.y*v3.y + v3.z*v3.z + v3.w*v3.w;
    const float inv = 1.0f / fmaxf(sqrtf(ss), EPS);

    if (copy_z) {
        *(float4*)(z + base + 0)  = v0;
        *(float4*)(z + base + 4)  = v1;
        *(float4*)(z + base + 8)  = v2;
        *(float4*)(z + base + 12) = v3;
    }

    v0.x *= inv; v0.y *= inv; v0.z *= inv; v0.w *= inv;
    v1.x *= inv; v1.y *= inv; v1.z *= inv; v1.w *= inv;
    v2.x *= inv; v2.y *= inv; v2.z *= inv; v2.w *= inv;
    v3.x *= inv; v3.y *= inv; v3.z *= inv; v3.w *= inv;

    *(float4*)(c + base + 0)  = v0;
    *(float4*)(c + base + 4)  = v1;
    *(float4*)(c + base + 8)  = v2;
    *(float4*)(c + base + 12) = v3;

    *(float4*)(agg + base + 0)  = v0;
    *(float4*)(agg + base + 4)  = v1;
    *(float4*)(agg + base + 8)  = v2;
    *(float4*)(agg + base + 12) = v3;
}

// ---------------------------------------------------------------------------
extern "C" void kernel_launch(void* const* d_in, const int* in_sizes, int n_in,
                              void* d_out, int out_size, void* d_ws, size_t ws_size,
                              hipStream_t stream)
{
    const float* X     = (const float*)d_in[0];   // [N, 256]
    const int*   edges = (const int*)  d_in[1];   // [2, M]
    const float* W     = (const float*)d_in[2];   // [8, 256, 16]
    const float* Bb    = (const float*)d_in[3];   // [8, 16]

    const int N = in_sizes[0] / INP;
    const int M = in_sizes[1] / 2;
    const int* src = edges;
    const int* trg = edges + M;

    float* c   = (float*)d_out;                       // [N,128] final output
    float* z   = (float*)d_ws;                        // [N,128]
    float* agg = z + (size_t)N * FDIM;                // [N,128]

    // init: z = c = agg = l2norm(relu(X W + b))
    disen_init_gemm<<<dim3((N + 15) / 16), 32, 0, stream>>>(X, W, Bb, z, c, agg, N);

    const int edge_blocks = (M + 7) / 8;              // 8 waves / block
    const int nk = N * KF;
    const int norm_blocks = (nk + 255) / 256;

    for (int t = 0; t < ROUTIT; ++t) {
        // agg already holds c on entry (seeded by gemm / previous renorm)
        disen_edge_route<<<edge_blocks, 256, 0, stream>>>(z, c, agg, src, trg, M);
        disen_renorm<<<norm_blocks, 256, 0, stream>>>(agg, c, z, t == 0 ? 1 : 0, nk);
    }
}